// Attention_68925635166686
// MI455X (gfx1250) — compile-verified
//
#include <hip/hip_runtime.h>

typedef __attribute__((ext_vector_type(2))) float v2f;
typedef __attribute__((ext_vector_type(4))) float v4f;
typedef __attribute__((ext_vector_type(8))) float v8f;

#define N_NODES 50000
#define D_DIM 64
#define ROW_STRIDE 68   // 64 + 4 pad: 16B-aligned rows, conflict-free banks

// ---------------- kernel 1: zero node accumulators ----------------
__global__ void k_init_nodes(float* __restrict__ node_max,
                             float* __restrict__ node_sum, int n) {
    int i = blockIdx.x * blockDim.x + threadIdx.x;
    if (i < n) { node_max[i] = 0.0f; node_sum[i] = 0.0f; }
}

// ---------------- kernel 2: per-edge score via f32 WMMA + segment max ----
// One wave handles a tile of 16 edges. score = diag(Qt * Kt^T) / 8.
__global__ void __launch_bounds__(128)
k_score_max(const float* __restrict__ q, const float* __restrict__ k,
            const int* __restrict__ index, float* __restrict__ score,
            float* __restrict__ node_max, int E) {
    __shared__ float lds[4 * 2 * 16 * ROW_STRIDE];
    const int lane = threadIdx.x & 31;
    const int wave = threadIdx.x >> 5;
    const int tile = blockIdx.x * 4 + wave;
    const int numTiles = (E + 15) >> 4;
    if (tile >= numTiles) return;   // wave-uniform: EXEC stays all-ones

    float* qs = &lds[wave * (2 * 16 * ROW_STRIDE)];
    float* ks = qs + 16 * ROW_STRIDE;

    // Stage 16x64 f32 Q and K tiles into LDS (coalesced 512B/instr).
    const long base = (long)tile * (16 * D_DIM);
    const long maxOff = (long)E * D_DIM - 4;
    #pragma unroll
    for (int i = 0; i < 8; ++i) {
        const int off = i * 128 + lane * 4;
        long g = base + off; if (g > maxOff) g = maxOff;   // tail clamp (stores masked later)
        const v4f qv = __builtin_nontemporal_load((const v4f*)(q + g));
        const v4f kv = __builtin_nontemporal_load((const v4f*)(k + g));
        const int row = off >> 6, col = off & 63;
        *(v4f*)(qs + row * ROW_STRIDE + col) = qv;
        *(v4f*)(ks + row * ROW_STRIDE + col) = kv;
    }

    // A (16x4 f32): lanes 0-15 hold K=c,c+1; lanes 16-31 hold K=c+2,c+3.
    // B = Kt^T (4x16): identical per-lane addressing (lane = edge column).
    const int row  = lane & 15;
    const int ksel = (lane >> 4) << 1;
    const float* qr = qs + row * ROW_STRIDE;
    const float* kr = ks + row * ROW_STRIDE;
    v8f acc = {0.f, 0.f, 0.f, 0.f, 0.f, 0.f, 0.f, 0.f};
    #pragma unroll
    for (int c = 0; c < 16; ++c) {
        const int k0 = (c << 2) + ksel;
        v2f a; a.x = qr[k0]; a.y = qr[k0 + 1];
        v2f b; b.x = kr[k0]; b.y = kr[k0 + 1];
        acc = __builtin_amdgcn_wmma_f32_16x16x4_f32(
                  false, a, false, b, (short)0, acc, false, false);
    }

    // Diagonal of C: M=N=m. m<8 -> lane m, vgpr m; m>=8 -> lane m+16, vgpr m-8.
    const bool have = (lane < 8) || (lane >= 24);
    const int  j    = (lane < 8) ? lane : (lane - 24);
    const int  edge = (lane < 8) ? lane : (lane - 16);
    float m0 = (j & 4) ? acc[4] : acc[0];
    float m1 = (j & 4) ? acc[5] : acc[1];
    float m2 = (j & 4) ? acc[6] : acc[2];
    float m3 = (j & 4) ? acc[7] : acc[3];
    float n0 = (j & 2) ? m2 : m0;
    float n1 = (j & 2) ? m3 : m1;
    float diag = (j & 1) ? n1 : n0;

    if (have) {
        const long e = (long)tile * 16 + edge;
        if (e < E) {
            const float s = diag * 0.125f;      // 1/sqrt(64)
            score[e] = s;
            // float-max via int atomicMax: valid since buffer init = +0.0
            // and the reference clamps the segment max at 0 (negatives never win).
            atomicMax((int*)(node_max + index[e]), __float_as_int(s));
        }
    }
}

// ---------------- kernel 3: ex = exp(s - smax), segment sum -------------
__global__ void k_exp_sum(const int* __restrict__ index,
                          const float* __restrict__ node_max,
                          float* __restrict__ score_ex,
                          float* __restrict__ node_sum, int E) {
    int e = blockIdx.x * blockDim.x + threadIdx.x;
    if (e >= E) return;
    const int n = index[e];
    const float ex = __expf(score_ex[e] - node_max[n]);
    score_ex[e] = ex;                      // in-place: score -> ex
    atomicAdd(node_sum + n, ex);
}

// ---------------- kernel 4: rdenom = 1/(sum + exp(-smax)) ---------------
__global__ void k_finalize(const float* __restrict__ node_max,
                           const float* __restrict__ node_sum,
                           float* __restrict__ rdenom, int n) {
    int i = blockIdx.x * blockDim.x + threadIdx.x;
    if (i < n) rdenom[i] = 1.0f / (node_sum[i] + __expf(-node_max[i]));
}

// ---------------- kernel 5: out = att * V (16B streaming) ---------------
__global__ void k_scale_v(const float* __restrict__ v,
                          const float* __restrict__ ex,
                          const int* __restrict__ index,
                          const float* __restrict__ rdenom,
                          float* __restrict__ out, int E) {
    const long gid = (long)blockIdx.x * blockDim.x + threadIdx.x;
    const long total = (long)E * (D_DIM / 4);
    if (gid >= total) return;
    const long e = gid >> 4;
    const float att = ex[e] * rdenom[index[e]];
    v4f vv = __builtin_nontemporal_load((const v4f*)v + gid);
    v4f o;
    o.x = att * vv.x; o.y = att * vv.y; o.z = att * vv.z; o.w = att * vv.w;
    __builtin_nontemporal_store(o, (v4f*)out + gid);
}

// ---------------- launch ------------------------------------------------
extern "C" void kernel_launch(void* const* d_in, const int* in_sizes, int n_in,
                              void* d_out, int out_size, void* d_ws, size_t ws_size,
                              hipStream_t stream) {
    const float* q     = (const float*)d_in[0];
    const float* k     = (const float*)d_in[1];
    const float* v     = (const float*)d_in[2];
    const int*   index = (const int*)d_in[3];
    const int E = in_sizes[3];
    const int N = N_NODES;

    float* ws       = (float*)d_ws;
    float* score    = ws;              // E floats (score, then ex in-place)
    float* node_max = ws + E;          // N floats
    float* node_sum = node_max + N;    // N floats
    float* rdenom   = node_sum + N;    // N floats
    float* out      = (float*)d_out;

    int nb = (N + 255) / 256;
    k_init_nodes<<<nb, 256, 0, stream>>>(node_max, node_sum, N);

    const int numTiles = (E + 15) / 16;
    nb = (numTiles + 3) / 4;
    k_score_max<<<nb, 128, 0, stream>>>(q, k, index, score, node_max, E);

    nb = (E + 255) / 256;
    k_exp_sum<<<nb, 256, 0, stream>>>(index, node_max, score, node_sum, E);

    nb = (N + 255) / 256;
    k_finalize<<<nb, 256, 0, stream>>>(node_max, node_sum, rdenom, N);

    const long total = (long)E * (D_DIM / 4);
    nb = (int)((total + 255) / 256);
    k_scale_v<<<nb, 256, 0, stream>>>(v, score, index, rdenom, out, E);
}